// Blender_34849364640269
// MI455X (gfx1250) — compile-verified
//
#include <hip/hip_runtime.h>
#include <math.h>

// Problem constants (from the reference): bases [1,4,136,200], box_feat [N,789],
// out [N,544,800] fp32.
#define NBASE 4
#define BH 136
#define BW 200
#define IMG_H 544
#define IMG_W 800
#define OUTD 56
#define CO 14
#define BF_STRIDE (5 + NBASE * CO * CO)   // 789

typedef float v2f __attribute__((ext_vector_type(2)));
typedef float v8f __attribute__((ext_vector_type(8)));

__device__ __forceinline__ float tentw(float d) {
    return fmaxf(0.0f, 1.0f - fabsf(d));
}

// -------------------------------------------------------------------------
// Kernel 1: fused roi_align + coefficient upsample + softmax + sigmoid
// One block per ROI; writes masks[n][56][56] (fp32) into workspace.
// -------------------------------------------------------------------------
__global__ __launch_bounds__(256) void blender_mask_kernel(
    const float* __restrict__ bases, const float* __restrict__ box_feat,
    float* __restrict__ masks) {
    const int n = blockIdx.x;
    const float* bf = box_feat + (size_t)n * BF_STRIDE;

    __shared__ float scoef[NBASE * CO * CO];
    __shared__ float sbox[4];
    for (int i = threadIdx.x; i < NBASE * CO * CO; i += 256) scoef[i] = bf[5 + i];
    if (threadIdx.x < 4) sbox[threadIdx.x] = bf[1 + threadIdx.x];
    __syncthreads();

    const int bidx = (int)bf[0];
    const float* feat = bases + (size_t)bidx * NBASE * BH * BW;

    // roi_align: s = box*0.25 - 0.5
    const float x0s = sbox[0] * 0.25f - 0.5f;
    const float y0s = sbox[1] * 0.25f - 0.5f;
    const float x1s = sbox[2] * 0.25f - 0.5f;
    const float y1s = sbox[3] * 0.25f - 0.5f;
    const float bwd = (x1s - x0s) / (float)OUTD;
    const float bhd = (y1s - y0s) / (float)OUTD;

    for (int p = threadIdx.x; p < OUTD * OUTD; p += 256) {
        const int py = p / OUTD;
        const int px = p - py * OUTD;

        // ---- roi_align sample (sampling_ratio=1, aligned=True) ----
        const float ys = y0s + ((float)py + 0.5f) * bhd;
        const float xs = x0s + ((float)px + 0.5f) * bwd;
        const bool valid = (ys >= -1.0f) && (ys <= (float)BH) &&
                           (xs >= -1.0f) && (xs <= (float)BW);
        const float yc = fminf(fmaxf(ys, 0.0f), (float)(BH - 1));
        const float xc = fminf(fmaxf(xs, 0.0f), (float)(BW - 1));
        const int yl = min(max((int)floorf(yc), 0), BH - 2);
        const int xl = min(max((int)floorf(xc), 0), BW - 2);
        const float ly = yc - (float)yl, hy = 1.0f - ly;
        const float lx = xc - (float)xl, hx = 1.0f - lx;

        // ---- coefficient upsample coords (jax.image.resize bilinear 14->56:
        //      half-pixel centers; weight renorm at the edges == clamp) ----
        float cyf = fminf(fmaxf(((float)py + 0.5f) * 0.25f - 0.5f, 0.0f), 13.0f);
        float cxf = fminf(fmaxf(((float)px + 0.5f) * 0.25f - 0.5f, 0.0f), 13.0f);
        const int cy = min((int)cyf, CO - 2);
        const int cx = min((int)cxf, CO - 2);
        const float fy2 = cyf - (float)cy, gy2 = 1.0f - fy2;
        const float fx2 = cxf - (float)cx, gx2 = 1.0f - fx2;

        float r[NBASE], w[NBASE];
        float wmax = -1e30f;
#pragma unroll
        for (int c = 0; c < NBASE; ++c) {
            const float* fc = feat + (size_t)c * BH * BW;
            const float f00 = fc[yl * BW + xl];
            const float f01 = fc[yl * BW + xl + 1];
            const float f10 = fc[(yl + 1) * BW + xl];
            const float f11 = fc[(yl + 1) * BW + xl + 1];
            float v = hy * (hx * f00 + lx * f01) + ly * (hx * f10 + lx * f11);
            r[c] = valid ? v : 0.0f;

            const float* cc = scoef + c * CO * CO;
            const float c00 = cc[cy * CO + cx];
            const float c01 = cc[cy * CO + cx + 1];
            const float c10 = cc[(cy + 1) * CO + cx];
            const float c11 = cc[(cy + 1) * CO + cx + 1];
            w[c] = gy2 * (gx2 * c00 + fx2 * c01) + fy2 * (gx2 * c10 + fx2 * c11);
            wmax = fmaxf(wmax, w[c]);
        }
        float se = 0.0f;
#pragma unroll
        for (int c = 0; c < NBASE; ++c) { w[c] = expf(w[c] - wmax); se += w[c]; }
        float acc = 0.0f;
#pragma unroll
        for (int c = 0; c < NBASE; ++c) acc += r[c] * w[c];
        acc /= se;
        masks[(size_t)n * OUTD * OUTD + p] = 1.0f / (1.0f + expf(-acc));
    }
}

// -------------------------------------------------------------------------
// Kernel 2: paste via WMMA.  out[n] = Wy (544x56) * masks (56x56) * Wx^T (56x800)
// One wave32 per (roi n, 16-wide column tile).  fp32 WMMA 16x16x4, K=56.
// -------------------------------------------------------------------------
__global__ __launch_bounds__(32) void blender_paste_kernel(
    const float* __restrict__ masks, const float* __restrict__ box_feat,
    float* __restrict__ out) {
    const int xt = blockIdx.x;           // 0..49
    const int n  = blockIdx.y;           // roi
    const int lane = threadIdx.x;        // 0..31
    const int q  = lane & 15;            // N index of the fragment
    const int hi = lane >> 4;            // half-lane group
    const int kb = hi * 2;               // K sub-offset for A/B fragments

    const float* bf = box_feat + (size_t)n * BF_STRIDE;
    const float x0 = bf[1], y0 = bf[2], x1 = bf[3], y1 = bf[4];
    const float sx = (float)OUTD / (x1 - x0);
    const float sy = (float)OUTD / (y1 - y0);
    const int xbase = xt * 16;
    float* outn = out + (size_t)n * IMG_H * IMG_W;

    // Column-tile tent support check (fx monotonic increasing in column).
    const float fx_lo = ((float)xbase + 0.5f - x0) * sx - 0.5f;
    const float fx_hi = ((float)xbase + 15.5f - x0) * sx - 0.5f;
    if (fx_hi <= -1.0f || fx_lo >= (float)OUTD) {
        for (int ht = 0; ht < IMG_H / 16; ++ht) {
            const int ybase = ht * 16;
#pragma unroll
            for (int r = 0; r < 8; ++r)
                outn[(size_t)(ybase + r + 8 * hi) * IMG_W + xbase + q] = 0.0f;
        }
        return;
    }

    __shared__ float smask[64 * OUTD];   // masks[n], zero-padded rows 56..63
    __shared__ float sC[64 * 16];        // C = masks * Wx^T  (column block)

    const float* mk = masks + (size_t)n * OUTD * OUTD;
    for (int i = lane; i < 64 * OUTD; i += 32)
        smask[i] = (i < OUTD * OUTD) ? mk[i] : 0.0f;
    __syncthreads();

    // per-lane Wx tent center for column q of this tile
    const float fxq = ((float)(xbase + q) + 0.5f - x0) * sx - 0.5f;

    // ---- C(64x16) = masks(64x56) @ Wx^T(56x16), four M-tiles, K-steps of 4 ----
    for (int mt = 0; mt < 4; ++mt) {
        v8f c = {};
#pragma unroll
        for (int k = 0; k < OUTD; k += 4) {
            // A fragment: masks[16*mt + q][k+kb], [k+kb+1]  (8B-aligned LDS pair)
            const float2 av = *(const float2*)&smask[(16 * mt + q) * OUTD + k + kb];
            v2f a; a[0] = av.x; a[1] = av.y;
            // B fragment: Wx^T[k+kb][q], [k+kb+1][q] = tent weights
            v2f b; b[0] = tentw(fxq - (float)(k + kb));
                   b[1] = tentw(fxq - (float)(k + kb + 1));
            c = __builtin_amdgcn_wmma_f32_16x16x4_f32(false, a, false, b,
                                                      (short)0, c, false, false);
        }
#pragma unroll
        for (int r = 0; r < 8; ++r)
            sC[(16 * mt + r + 8 * hi) * 16 + q] = c[r];
    }
    __syncthreads();

    // ---- out tiles: D(16x16) = Wy_tile(16x56) @ C(56x16) ----
    for (int ht = 0; ht < IMG_H / 16; ++ht) {
        const int ybase = ht * 16;
        const float fy_lo = ((float)ybase + 0.5f - y0) * sy - 0.5f;
        const float fy_hi = ((float)ybase + 15.5f - y0) * sy - 0.5f;
        if (fy_hi <= -1.0f || fy_lo >= (float)OUTD) {
#pragma unroll
            for (int r = 0; r < 8; ++r)
                outn[(size_t)(ybase + r + 8 * hi) * IMG_W + xbase + q] = 0.0f;
            continue;
        }
        const float fyh = ((float)(ybase + q) + 0.5f - y0) * sy - 0.5f;
        v8f d = {};
#pragma unroll
        for (int k = 0; k < OUTD; k += 4) {
            // A fragment: Wy[ybase + q][k+kb], [k+kb+1]
            v2f a; a[0] = tentw(fyh - (float)(k + kb));
                   a[1] = tentw(fyh - (float)(k + kb + 1));
            // B fragment: C[k+kb][q], C[k+kb+1][q]
            v2f b; b[0] = sC[(k + kb) * 16 + q];
                   b[1] = sC[(k + kb + 1) * 16 + q];
            d = __builtin_amdgcn_wmma_f32_16x16x4_f32(false, a, false, b,
                                                      (short)0, d, false, false);
        }
#pragma unroll
        for (int r = 0; r < 8; ++r)
            outn[(size_t)(ybase + r + 8 * hi) * IMG_W + xbase + q] = d[r];
    }
}

// -------------------------------------------------------------------------
extern "C" void kernel_launch(void* const* d_in, const int* in_sizes, int n_in,
                              void* d_out, int out_size, void* d_ws, size_t ws_size,
                              hipStream_t stream) {
    const float* bases    = (const float*)d_in[0];
    const float* box_feat = (const float*)d_in[1];
    float* out   = (float*)d_out;
    float* masks = (float*)d_ws;                 // N*56*56 fp32 = 1.25 MB

    const int N = in_sizes[1] / BF_STRIDE;       // 100

    blender_mask_kernel<<<dim3(N), dim3(256), 0, stream>>>(bases, box_feat, masks);
    blender_paste_kernel<<<dim3(IMG_W / 16, N), dim3(32), 0, stream>>>(masks, box_feat, out);
}